// HashGrid2D_43482248905254
// MI455X (gfx1250) — compile-verified
//
#include <hip/hip_runtime.h>
#include <cmath>

// ---------------------------------------------------------------------------
// InstantNGP-style 2D hash-grid interpolation for MI455X (gfx1250).
//
// Memory-bound gather kernel. Feature table (2 x 16 x 2^19 x 2 fp32 = 134 MB)
// fits in the 192 MB global L2 -> keep gathers cacheable (RT), but stream the
// 128 MB output with non-temporal stores so it does not evict the table.
//
// All gather addresses are 32-bit element indices into the table
// (max index 2^24-1), so every gather uses the GVS addressing mode:
//   global_load_b64 vdst, v_idx, s[features] scale_offset
// i.e. SGPR base + u32 VGPR offset scaled x8 — no per-lane 64-bit math.
// ---------------------------------------------------------------------------

#define N_LEVELS 16
#define LOG2_T   19
#define T_MASK   ((1u << LOG2_T) - 1u)
#define PRIME_Y  2654435761u

typedef float v2f __attribute__((ext_vector_type(2)));
typedef float v4f __attribute__((ext_vector_type(4)));

struct ScaleArgs { float s[N_LEVELS]; };   // (res_level - 1) as float

__global__ __launch_bounds__(256) void hashgrid2d_kernel(
    const v2f* __restrict__ x,          // (N, 2) fp32
    const v2f* __restrict__ features,   // (2, 16, 2^19, 2) fp32, viewed as v2f entries
    const int* __restrict__ hash_idxs,  // (N, 1) int32 in {0,1}
    v4f*       __restrict__ out,        // (N, 32) fp32, viewed as N*8 v4f
    int N, ScaleArgs sc)
{
    const int n = blockIdx.x * blockDim.x + threadIdx.x;
    if (n >= N) return;

    // Streamed-once inputs: non-temporal so they don't pollute L2.
    const v2f p = __builtin_nontemporal_load(&x[n]);
    const int g = __builtin_nontemporal_load(&hash_idxs[n]);

    // 32-bit element-index base of this hash-grid's sub-table: g * 16 * 2^19.
    const unsigned gbase = (unsigned)g << (4 + LOG2_T);

    v4f acc[8];

#pragma unroll
    for (int l = 0; l < N_LEVELS; ++l) {
        const float s  = sc.s[l];                 // compile-time field index -> SGPR
        const float px = p.x * s;
        const float py = p.y * s;
        const float fx = floorf(px);
        const float fy = floorf(py);
        const float wx = px - fx;
        const float wy = py - fy;

        const unsigned ix = (unsigned)(int)fx;
        const unsigned iy = (unsigned)(int)fy;

        // h(cx, cy) = (cx ^ cy*PRIME) & (T-1); reuse iy*PRIME for the y+1 row.
        const unsigned a0 = iy * PRIME_Y;
        const unsigned a1 = a0 + PRIME_Y;

        // Per-level sub-table base (low 19 bits zero) -> (hash & MASK) | lb
        // fuses to v_and_or_b32 after the v_xor.
        const unsigned lb = gbase + ((unsigned)l << LOG2_T);

        const unsigned i00 = (( ix        ^ a0) & T_MASK) | lb;
        const unsigned i10 = (((ix + 1u)  ^ a0) & T_MASK) | lb;
        const unsigned i01 = (( ix        ^ a1) & T_MASK) | lb;
        const unsigned i11 = (((ix + 1u)  ^ a1) & T_MASK) | lb;

        // Four independent 8-byte gathers, default RT hint: table lines are
        // reused ~30x on average and stay resident in the 192 MB L2.
        const v2f f00 = features[i00];
        const v2f f10 = features[i10];
        const v2f f01 = features[i01];
        const v2f f11 = features[i11];

        const float omx = 1.0f - wx;
        const float omy = 1.0f - wy;
        const float w00 = omx * omy;
        const float w10 = wx  * omy;
        const float w01 = omx * wy;
        const float w11 = wx  * wy;

        const float o0 = fmaf(w11, f11.x, fmaf(w01, f01.x, fmaf(w10, f10.x, w00 * f00.x)));
        const float o1 = fmaf(w11, f11.y, fmaf(w01, f01.y, fmaf(w10, f10.y, w00 * f00.y)));

        if ((l & 1) == 0) { acc[l >> 1].x = o0; acc[l >> 1].y = o1; }
        else              { acc[l >> 1].z = o0; acc[l >> 1].w = o1; }
    }

    // 128 B of output per point: 8 x global_store_b128 with TH_NT so the
    // 128 MB output stream does not evict the 134 MB feature table from L2.
    v4f* __restrict__ o = out + (size_t)n * 8;
#pragma unroll
    for (int k = 0; k < 8; ++k)
        __builtin_nontemporal_store(acc[k], &o[k]);
}

extern "C" void kernel_launch(void* const* d_in, const int* in_sizes, int n_in,
                              void* d_out, int out_size, void* d_ws, size_t ws_size,
                              hipStream_t stream) {
    (void)n_in; (void)out_size; (void)d_ws; (void)ws_size;

    const v2f* x         = (const v2f*)d_in[0];   // (N,2) fp32
    const v2f* features  = (const v2f*)d_in[1];   // (2,16,T,2) fp32
    const int* hash_idxs = (const int*)d_in[2];   // (N,1) int32
    v4f*       out       = (v4f*)d_out;           // (N,32) fp32

    const int N = in_sizes[0] / 2;

    // Mirror the reference's double-precision resolution computation exactly
    // (math.exp/log and float**int all route through libm pow/exp/log, and
    // int() truncates toward zero).
    ScaleArgs sc;
    const double b = exp((log(512.0) - log(16.0)) / (double)(N_LEVELS - 1));
    for (int i = 0; i < N_LEVELS; ++i) {
        const int res = (int)(16.0 * pow(b, (double)i));
        sc.s[i] = (float)(res - 1);
    }

    const int block = 256;                 // 8 wave32s per block
    const int grid  = (N + block - 1) / block;
    hipLaunchKernelGGL(hashgrid2d_kernel, dim3(grid), dim3(block), 0, stream,
                       x, features, hash_idxs, out, N, sc);
}